// TemporalLayer_12953621364818
// MI455X (gfx1250) — compile-verified
//
#include <hip/hip_runtime.h>
#include <cstdint>

// ---------------------------------------------------------------------------
// Periodic 5-point stencil, lattice 512x512, batch 256.
// out = (1-4D)*x + (D-0.5v1)*right + (D+0.5v1)*left
//       + (D-0.5v2)*up + (D+0.5v2)*down + b_f,  D = d*d
// Memory-bound: ~0.54 GB traffic -> ~23 us floor @ 23.3 TB/s.
// One 128-thread block per (sample,row); float4 per thread.
// Center row staged to LDS via CDNA5 async global->LDS B128 (ASYNCcnt),
// up/down rows loaded straight to VGPRs as B128, NT stores for output.
// ---------------------------------------------------------------------------

#define LAT 512

#if __has_builtin(__builtin_amdgcn_global_load_async_to_lds_b128) && \
    __has_builtin(__builtin_amdgcn_s_wait_asynccnt)
#define USE_ASYNC_LDS 1
#else
#define USE_ASYNC_LDS 0
#warning "gfx1250 async-to-LDS builtins NOT available on this toolchain; using plain LDS staging"
#endif

typedef float float4v __attribute__((ext_vector_type(4)));
typedef int   v4i     __attribute__((ext_vector_type(4)));
typedef __attribute__((address_space(1))) v4i g_v4i;   // global
typedef __attribute__((address_space(3))) v4i l_v4i;   // LDS

__global__ __launch_bounds__(128) void stencil512_kernel(
    const float* __restrict__ x,
    const float* __restrict__ dP,
    const float* __restrict__ v1P,
    const float* __restrict__ v2P,
    const float* __restrict__ bfP,
    float* __restrict__ out)
{
    __shared__ __align__(16) float srow[LAT];

    const int tid = threadIdx.x;          // 0..127, each owns 4 columns
    const int row = blockIdx.x;           // 0..511
    const int b   = blockIdx.y;           // sample

    const long base  = (long)b * (long)(LAT * LAT);
    const int rowUp  = (row + LAT - 1) & (LAT - 1);
    const int rowDn  = (row + 1) & (LAT - 1);

    const float* gC = x + base + (long)row   * LAT;
    const float* gU = x + base + (long)rowUp * LAT;
    const float* gD = x + base + (long)rowDn * LAT;

    const int c0i = tid * 4;

    // ---- stage center row into LDS (async DMA path on CDNA5) ----
#if USE_ASYNC_LDS
    __builtin_amdgcn_global_load_async_to_lds_b128(
        (g_v4i*)(gC + c0i),
        (l_v4i*)(&srow[c0i]),
        0, 0);
#else
    *(float4v*)&srow[c0i] = *(const float4v*)(gC + c0i);
#endif

    // ---- up/down rows straight to registers (overlap with async copy) ----
    const float4v u  = *(const float4v*)(gU + c0i);
    const float4v dn = *(const float4v*)(gD + c0i);

    // ---- uniform scalar params (scalar loads) ----
    const float dd  = dP[0];
    const float v1  = v1P[0];
    const float v2  = v2P[0];
    const float bf  = bfP[0];
    const float D   = dd * dd;
    const float c0  = 1.0f - 4.0f * D;
    const float cR  = D - 0.5f * v1;
    const float cL  = D + 0.5f * v1;
    const float cU  = D - 0.5f * v2;
    const float cDn = D + 0.5f * v2;

#if USE_ASYNC_LDS
    __builtin_amdgcn_s_wait_asynccnt(0);   // my wave's async LDS writes landed
#endif
    __syncthreads();                        // everyone's row segment visible

    // ---- compute: vector center read + two halo scalars from LDS ----
    const float4v ce = *(const float4v*)&srow[c0i];
    const float lfS  = srow[(c0i + LAT - 1) & (LAT - 1)];
    const float rtS  = srow[(c0i + 4)       & (LAT - 1)];

    float4v r;
    {
        // j = 0
        r.x = fmaf(c0, ce.x, fmaf(cR, ce.y, fmaf(cL, lfS,
              fmaf(cU, u.x, fmaf(cDn, dn.x, bf)))));
        // j = 1
        r.y = fmaf(c0, ce.y, fmaf(cR, ce.z, fmaf(cL, ce.x,
              fmaf(cU, u.y, fmaf(cDn, dn.y, bf)))));
        // j = 2
        r.z = fmaf(c0, ce.z, fmaf(cR, ce.w, fmaf(cL, ce.y,
              fmaf(cU, u.z, fmaf(cDn, dn.z, bf)))));
        // j = 3
        r.w = fmaf(c0, ce.w, fmaf(cR, rtS, fmaf(cL, ce.z,
              fmaf(cU, u.w, fmaf(cDn, dn.w, bf)))));
    }

    // output written once, never re-read -> non-temporal store keeps L2
    // for the 3x-reused input rows
    __builtin_nontemporal_store(r, (float4v*)(out + base + (long)row * LAT + c0i));
}

extern "C" void kernel_launch(void* const* d_in, const int* in_sizes, int n_in,
                              void* d_out, int out_size, void* d_ws, size_t ws_size,
                              hipStream_t stream) {
    (void)n_in; (void)out_size; (void)d_ws; (void)ws_size;

    const float* x   = (const float*)d_in[0];
    const float* dP  = (const float*)d_in[1];
    const float* v1P = (const float*)d_in[2];
    const float* v2P = (const float*)d_in[3];
    const float* bfP = (const float*)d_in[4];
    float* outP      = (float*)d_out;

    const int B = in_sizes[0] / (LAT * LAT);   // 256 samples

    dim3 grid(LAT, B);      // one block per (row, sample)
    dim3 block(128);        // 4 wave32s, float4 per lane = 512 floats/row
    stencil512_kernel<<<grid, block, 0, stream>>>(x, dP, v1P, v2P, bfP, outP);
}